// CompressDCT_22703197127203
// MI455X (gfx1250) — compile-verified
//
#include <hip/hip_runtime.h>
#include <hip/hip_bf16.h>

typedef float v2f __attribute__((ext_vector_type(2)));
typedef float v8f __attribute__((ext_vector_type(8)));

// 8x8 DCT-II matrix (row 0 scaled by 1/sqrt(2)), matches reference _dct_matrix().
__device__ __constant__ float DCT8[64] = {
  0.35355339f,  0.35355339f,  0.35355339f,  0.35355339f,  0.35355339f,  0.35355339f,  0.35355339f,  0.35355339f,
  0.49039264f,  0.41573481f,  0.27778512f,  0.09754516f, -0.09754516f, -0.27778512f, -0.41573481f, -0.49039264f,
  0.46193977f,  0.19134172f, -0.19134172f, -0.46193977f, -0.46193977f, -0.19134172f,  0.19134172f,  0.46193977f,
  0.41573481f, -0.09754516f, -0.49039264f, -0.27778512f,  0.27778512f,  0.49039264f,  0.09754516f, -0.41573481f,
  0.35355339f, -0.35355339f, -0.35355339f,  0.35355339f,  0.35355339f, -0.35355339f, -0.35355339f,  0.35355339f,
  0.27778512f, -0.49039264f,  0.09754516f,  0.41573481f, -0.41573481f, -0.09754516f,  0.49039264f, -0.27778512f,
  0.19134172f, -0.46193977f,  0.46193977f, -0.19134172f, -0.19134172f,  0.46193977f, -0.46193977f,  0.19134172f,
  0.09754516f, -0.27778512f,  0.41573481f, -0.49039264f,  0.49039264f, -0.41573481f,  0.27778512f, -0.09754516f
};

// Block-diagonal 16x16: Dbig = diag(DCT8, DCT8)
__device__ __forceinline__ float dbig(int r, int c) {
  return ((r ^ c) & 8) ? 0.0f : DCT8[(r & 7) * 8 + (c & 7)];
}

#define IMG_W 128
#define WAVES_PER_BLOCK 8

__global__ __launch_bounds__(256) void dct8x8_wmma_kernel(
    const float* __restrict__ x, const float* __restrict__ qt,
    float* __restrict__ out, int iters, int total_waves) {
  const int lane = threadIdx.x & 31;
  const int wave = threadIdx.x >> 5;
  const int half = lane >> 4;   // which 16-lane half
  const int l16  = lane & 15;

  __shared__ float ldsX[WAVES_PER_BLOCK][16][16];  // tile staging (b128-friendly)
  __shared__ float ldsT[WAVES_PER_BLOCK][16][17];  // C/D -> A relayout, pad 17 = conflict-free

  // Constant Dbig slices in A layout (also serve as B operand of GEMM2, since
  // B[k][n] = Dbig[n][k] lands on identical lane/VGPR positions).
  v2f dsl[4];
#pragma unroll
  for (int s = 0; s < 4; ++s) {
    const int kb = 4 * s + 2 * half;
    dsl[s].x = dbig(l16, kb);
    dsl[s].y = dbig(l16, kb + 1);
  }
  // Per-lane quantization reciprocal column: q[i][n%8], i = C/D VGPR index.
  float rq[8];
#pragma unroll
  for (int i = 0; i < 8; ++i) rq[i] = 1.0f / qt[i * 8 + (l16 & 7)];

  const int wid = blockIdx.x * WAVES_PER_BLOCK + wave;

  for (int it = 0; it < iters; ++it) {
    const int tile = wid + it * total_waves;   // uniform trip count: no divergence, EXEC all-1s
    const int img = tile >> 6;                 // 64 tiles (8x8 of 16x16) per 128x128 image
    const int tr  = (tile >> 3) & 7;
    const int tc  = tile & 7;
    const size_t base = (size_t)img * (IMG_W * IMG_W) + (size_t)(tr * 16) * IMG_W + tc * 16;
    const float* __restrict__ src = x + base;
    float* __restrict__ dst = out + base;

    // ---- Load 16x16 f32 tile: 2 x global_load_b128 per lane -> LDS ----
#pragma unroll
    for (int p = 0; p < 2; ++p) {
      const int chunk = p * 32 + lane;         // 64 chunks of 16B
      const int r  = chunk >> 2;
      const int c4 = (chunk & 3) << 2;
      const float4 v = *reinterpret_cast<const float4*>(src + (size_t)r * IMG_W + c4);
      *reinterpret_cast<float4*>(&ldsX[wave][r][c4]) = v;
    }
    __syncthreads();

    // ---- GEMM1: T = Dbig(16x16) * X(16x16), four K=4 slices ----
    v8f acc = {0.f, 0.f, 0.f, 0.f, 0.f, 0.f, 0.f, 0.f};
#pragma unroll
    for (int s = 0; s < 4; ++s) {
      const int kb = 4 * s + 2 * half;         // B layout: lane=N, VGPRs hold successive K
      v2f b;
      b.x = ldsX[wave][kb][l16];
      b.y = ldsX[wave][kb + 1][l16];
      acc = __builtin_amdgcn_wmma_f32_16x16x4_f32(
          false, dsl[s], false, b, (short)0, acc, false, false);
    }

    // ---- Relayout T (C/D layout) through LDS into A layout ----
#pragma unroll
    for (int vi = 0; vi < 8; ++vi)
      ldsT[wave][vi + 8 * half][l16] = acc[vi];
    __syncthreads();

    // ---- GEMM2: Y = T * Dbig^T ----
    v8f acc2 = {0.f, 0.f, 0.f, 0.f, 0.f, 0.f, 0.f, 0.f};
#pragma unroll
    for (int s = 0; s < 4; ++s) {
      const int kb = 4 * s + 2 * half;
      v2f a;
      a.x = ldsT[wave][l16][kb];
      a.y = ldsT[wave][l16][kb + 1];
      acc2 = __builtin_amdgcn_wmma_f32_16x16x4_f32(
          false, a, false, dsl[s], (short)0, acc2, false, false);
    }

    // ---- Epilogue: round(Y/q), clamp [-128,127], store from C/D layout ----
#pragma unroll
    for (int vi = 0; vi < 8; ++vi) {
      const int m = vi + 8 * half;             // C/D: VGPR vi -> row m, col l16
      float v = rintf(acc2[vi] * rq[vi]);
      v = fminf(127.0f, fmaxf(-128.0f, v));
      dst[(size_t)m * IMG_W + l16] = v;
    }
  }
}

extern "C" void kernel_launch(void* const* d_in, const int* in_sizes, int n_in,
                              void* d_out, int out_size, void* d_ws, size_t ws_size,
                              hipStream_t stream) {
  const float* x  = (const float*)d_in[0];
  const float* qt = (const float*)d_in[1];
  float* out = (float*)d_out;

  const int n_elem = in_sizes[0];              // 32*64*128*128 = 33,554,432
  const int tiles  = n_elem / 256;             // 16x16 tiles = 131,072

  int blocks = 2048;                           // 16,384 persistent waves
  int total_waves = blocks * WAVES_PER_BLOCK;
  int iters = tiles / total_waves;             // 8 (exact for reference shape)
  if (iters * total_waves != tiles) {          // generic fallback: 1 tile per wave
    blocks = (tiles + WAVES_PER_BLOCK - 1) / WAVES_PER_BLOCK;
    total_waves = blocks * WAVES_PER_BLOCK;
    iters = 1;
  }

  dct8x8_wmma_kernel<<<blocks, 256, 0, stream>>>(x, qt, out, iters, total_waves);
}